// Render_50792283242842
// MI455X (gfx1250) — compile-verified
//
#include <hip/hip_runtime.h>

// ---------------------------------------------------------------------------
// Point-splat renderer for MI455X (gfx1250, wave32).
// Pipeline: init -> z scatter-min -> row 5-tap min -> TDM-tiled col 5-tap min
//           -> 7x7 splat with WMMA-computed distance field.
// CDNA5 paths: v_wmma_f32_16x16x4_f32 (distance matrix = A[49x4] x B[4x16]),
//              tensor_load_to_lds + s_wait_tensorcnt (TDM 2D tile DMA),
//              global_prefetch_b8, global_atomic_{add_f32,min_u32}, LDS.
// ---------------------------------------------------------------------------

#define HH 576
#define WW 640
#define HWSZ (HH * WW)

// d_out layout (floats), reference tuple order:
//   depth[HW] | color[HW*3] | imw[HW] | weight[HW] | is_visible[N]
#define OFF_DEPTH 0
#define OFF_COLOR (HWSZ)
#define OFF_IMW   (HWSZ * 4)
#define OFF_WT    (HWSZ * 5)
#define OFF_VIS   (HWSZ * 6)

#define EPSF 1e-5f

// ---- TDM tile geometry for the vertical min pass ----
#define TS_X   64                    // tile width  (WW/TS_X = 10, exact)
#define TS_Y   16                    // tile height (HH/TS_Y = 36, exact)
#define THALO  2                     // +/-2 rows for the 5-tap window
#define TROWS  (TS_Y + 2 * THALO)    // 20 rows staged in LDS

typedef __attribute__((ext_vector_type(4))) unsigned int u32x4;
typedef __attribute__((ext_vector_type(8))) int          i32x8;
typedef __attribute__((ext_vector_type(4))) int          i32x4;
typedef __attribute__((ext_vector_type(2))) float        v2f;
typedef __attribute__((ext_vector_type(8))) float        v8f;

// ---------------------------------------------------------------------------
// K1: depth workspace <- BIG, accumulation regions of d_out <- 0
// ---------------------------------------------------------------------------
__global__ void k_init(float* __restrict__ dws, float* __restrict__ out) {
    int i = blockIdx.x * blockDim.x + threadIdx.x;
    if (i < HWSZ) dws[i] = 1e10f;
    if (i < 5 * HWSZ) out[OFF_COLOR + i] = 0.0f;   // color+imw+weight = 5*HW
}

// ---------------------------------------------------------------------------
// K2: per-pixel min depth via uint atomic-min (positive floats: uint order ==
// float order, so this is an exact float scatter-min in L2).
// ---------------------------------------------------------------------------
__global__ void k_zmin(const float* __restrict__ pts,
                       const int* __restrict__ mask,
                       unsigned int* __restrict__ dws_u, int N) {
    int i = blockIdx.x * blockDim.x + threadIdx.x;
    if (i >= N) return;
    float x = pts[3 * i + 0];
    float y = pts[3 * i + 1];
    float z = pts[3 * i + 2];
    int px = (int)floorf(x);
    int py = (int)floorf(y);
    bool valid = (px >= 0) & (px < WW) & (py >= 0) & (py < HH) & (mask[i] > 0);
    if (valid) {
        atomicMin(&dws_u[py * WW + px], __float_as_uint(z));
    }
}

// ---------------------------------------------------------------------------
// K3a: horizontal 5-tap min; also finalize depth image (BIG -> 0).
// ---------------------------------------------------------------------------
__global__ void k_rowmin(const float* __restrict__ dws,
                         float* __restrict__ rmin,
                         float* __restrict__ out) {
    int i = blockIdx.x * blockDim.x + threadIdx.x;
    if (i >= HWSZ) return;
    int x = i % WW;
    float m = 1e10f;
#pragma unroll
    for (int dx = -2; dx <= 2; ++dx) {
        int xx = x + dx;
        if (xx >= 0 && xx < WW) m = fminf(m, dws[i + dx]);
    }
    rmin[i] = m;
    float d = dws[i];
    out[OFF_DEPTH + i] = (d >= 1e10f) ? 0.0f : d;   // depth_image finalize
}

// ---------------------------------------------------------------------------
// K3b: vertical 5-tap min, tile staged into LDS by the Tensor Data Mover.
// ---------------------------------------------------------------------------
__global__ void __launch_bounds__(256)
k_colmin_tdm(const float* __restrict__ rmin, float* __restrict__ pmin) {
    __shared__ float tile[TROWS * TS_X];          // 5 KB of the 320 KB LDS

    const int bx  = blockIdx.x % (WW / TS_X);
    const int by  = blockIdx.x / (WW / TS_X);
    const int gx0 = bx * TS_X;
    const int gy0 = by * TS_Y;

    // 1) pre-fill with BIG so out-of-image halo rows act as padding
    for (int l = threadIdx.x; l < TROWS * TS_X; l += 256)
        tile[l] = 1e10f;
    __syncthreads();

    // 2) wave 0 issues the TDM tile load of the in-bounds row range
    const int ys       = max(gy0 - THALO, 0);
    const int ye       = min(gy0 + TS_Y + THALO - 1, HH - 1);
    const int nrows    = ye - ys + 1;             // 18 (edges) or 20
    const int lds_row0 = ys - (gy0 - THALO);      // rows skipped at top

    if ((threadIdx.x >> 5) == 0) {                // wave-uniform branch
        unsigned long long gaddr =
            (unsigned long long)(uintptr_t)(rmin + (size_t)ys * WW + gx0);
        unsigned int laddr =
            (unsigned int)(uintptr_t)(&tile[lds_row0 * TS_X]);

        // D# group 0: count=1 | lds_addr | global_addr[56:0] | type=2
        u32x4 g0;
        g0.x = 1u;
        g0.y = laddr;
        g0.z = (unsigned int)(gaddr & 0xffffffffu);
        g0.w = (unsigned int)((gaddr >> 32) & 0x1ffffffu) | (2u << 30);

        // D# group 1: data_size=4B; tensor 640x576 stride 640; tile 64 x nrows
        i32x8 g1;
        g1[0] = (2 << 16);                                    // data_size=2 (4B)
        g1[1] = (int)(((unsigned)WW & 0xffffu) << 16);        // tdim0 lo16
        g1[2] = (int)(((unsigned)WW >> 16) |
                      (((unsigned)HH & 0xffffu) << 16));      // tdim0 hi|tdim1 lo
        g1[3] = (int)(((unsigned)HH >> 16) |
                      ((unsigned)TS_X << 16));                // tdim1 hi|tile_dim0
        g1[4] = nrows & 0xffff;                               // tile_dim1 (dim2=0)
        g1[5] = WW;                                           // dim0_stride lo32
        g1[6] = 0;                                            // stride hi | d1s lo
        g1[7] = 0;                                            // d1s hi (2D: unused)

        i32x4 z4 = {0, 0, 0, 0};                              // groups 2/3 unused
        i32x8 z8 = {0, 0, 0, 0, 0, 0, 0, 0};                  // 6-arg form pad
        __builtin_amdgcn_tensor_load_to_lds(g0, g1, z4, z4, z8, 0);
        __builtin_amdgcn_s_wait_tensorcnt(0);                 // TENSORcnt==0
    }
    __syncthreads();                                          // publish LDS tile

    // 3) 5-tap vertical min out of LDS; each thread covers 4 output rows
    const int tx  = threadIdx.x & (TS_X - 1);
    const int ty0 = threadIdx.x / TS_X;                       // 0..3
#pragma unroll
    for (int r = ty0; r < TS_Y; r += 4) {
        float m =      tile[(r + 0) * TS_X + tx];
        m = fminf(m,   tile[(r + 1) * TS_X + tx]);
        m = fminf(m,   tile[(r + 2) * TS_X + tx]);
        m = fminf(m,   tile[(r + 3) * TS_X + tx]);
        m = fminf(m,   tile[(r + 4) * TS_X + tx]);
        pmin[(gy0 + r) * WW + gx0 + tx] = m;
    }
}

// ---------------------------------------------------------------------------
// K4: visibility + 7x7 splat. Distance field computed on the matrix pipe:
//   d[tap, pt] = 2ox*a + 2oy*b + 1*(a^2+b^2) + (ox^2+oy^2+eps)
// as D = A[16x4] x B[4x16] per tap-group via v_wmma_f32_16x16x4_f32.
// C/D layout: lane l (0..15) = point column l rows 0..7; lane l+16 = rows
// 8..15 of the same point, so a lane PAIR splits one point's 49 taps.
// N % 32 == 0 and the early-out is wave-uniform -> EXEC all-ones at WMMA.
// ---------------------------------------------------------------------------
__global__ void __launch_bounds__(256)
k_splat_wmma(const float* __restrict__ pts,
             const float* __restrict__ col,
             const float* __restrict__ imw,
             const int* __restrict__ mask,
             const float* __restrict__ th,
             const float* __restrict__ pmin,
             float* __restrict__ out, int N) {
    const int lane = threadIdx.x & 31;
    const int i    = blockIdx.x * blockDim.x + threadIdx.x;
    if ((i - lane) >= N) return;                  // wave-uniform early out

    const bool inr = (i < N);
    if (inr) {
        __builtin_prefetch(&col[3 * i], 0, 0);    // -> global_prefetch_b8
        __builtin_prefetch(&imw[i], 0, 0);
    }

    float x = 0.f, y = 0.f, z = 0.f; int m0 = 0;
    if (inr) {
        x = pts[3 * i + 0]; y = pts[3 * i + 1]; z = pts[3 * i + 2];
        m0 = mask[i];
    }
    int px = (int)floorf(x);
    int py = (int)floorf(y);
    bool valid = (px >= 0) & (px < WW) & (py >= 0) & (py < HH) & (m0 > 0);

    int cx = min(max(px, 0), WW - 1);
    int cy = min(max(py, 0), HH - 1);
    float pm = pmin[cy * WW + cx];
    bool vis = inr && valid && (z <= pm + th[0]);
    if (inr) out[OFF_VIS + i] = vis ? 1.0f : 0.0f;

    float c0 = 0.f, c1 = 0.f, c2 = 0.f, w0 = 0.f;
    if (vis) {
        c0 = col[3 * i + 0]; c1 = col[3 * i + 1]; c2 = col[3 * i + 2];
        w0 = imw[i];
    }

    // Sub-pixel offsets: dx(tap) = ox + a, dy(tap) = oy + b
    float a = (float)px + 0.5f - x;
    float b = (float)py + 0.5f - y;
    float s = a * a + b * b;

    const bool hi = (lane >= 16);
    const int  ml = lane & 15;                    // A/B row|col index

    // Two 16-point batches per wave
    for (int t = 0; t < 2; ++t) {
        const int src = ml | (t << 4);            // owner lane of point column
        float av  = __shfl(a,  src, 32);
        float bv  = __shfl(b,  src, 32);
        float sv  = __shfl(s,  src, 32);
        int   pxv = __shfl(px, src, 32);
        int   pyv = __shfl(py, src, 32);
        float c0v = __shfl(c0, src, 32);
        float c1v = __shfl(c1, src, 32);
        float c2v = __shfl(c2, src, 32);
        float w0v = __shfl(w0, src, 32);
        int  visv = __shfl((int)vis, src, 32);

        // B[4x16]: K0=a, K1=b, K2=a^2+b^2, K3=1  (lanes 0-15: K0/K1, 16-31: K2/K3)
        v2f B;
        B.x = hi ? sv   : av;
        B.y = hi ? 1.0f : bv;

        v8f D[4];
#pragma unroll
        for (int g = 0; g < 4; ++g) {
            // A[16x4] tap-group g: row m -> tap = g*16+m
            int tap = g * 16 + ml;
            int oyi = tap / 7 - 3;
            int oxi = tap - (tap / 7) * 7 - 3;
            bool live = (tap < 49);
            float fox = (float)oxi, foy = (float)oyi;
            v2f A;
            if (!hi) {                            // K0 = 2ox, K1 = 2oy
                A.x = live ? 2.0f * fox : 0.0f;
                A.y = live ? 2.0f * foy : 0.0f;
            } else {                              // K2 = 1, K3 = ox^2+oy^2+eps
                A.x = live ? 1.0f : 0.0f;
                A.y = live ? (fox * fox + foy * foy + EPSF) : 0.0f;
            }
            v8f cz = {};
            D[g] = __builtin_amdgcn_wmma_f32_16x16x4_f32(
                false, A, false, B, (short)0, cz, false, false);
        }

        if (visv) {
            const int hioff = hi ? 8 : 0;         // lo lane: rows 0-7, hi: 8-15
#pragma unroll
            for (int g = 0; g < 4; ++g) {
#pragma unroll
                for (int r = 0; r < 8; ++r) {
                    int tap = g * 16 + r + hioff;
                    if (tap >= 49) continue;      // pad slots
                    int oyi = tap / 7 - 3;
                    int oxi = tap - (tap / 7) * 7 - 3;
                    int sy = pyv + oyi;
                    int sx = pxv + oxi;
                    if (sx < 0 || sx >= WW || sy < 0 || sy >= HH) continue;
                    // d > eps > 0; rcp of exact-1 numerator == IEEE division
                    float w = __frcp_rn(D[g][r]);
                    int p = sy * WW + sx;
                    atomicAdd(&out[OFF_WT + p], w);                  // weight
                    atomicAdd(&out[OFF_COLOR + 3 * p + 0], w * c0v); // color
                    atomicAdd(&out[OFF_COLOR + 3 * p + 1], w * c1v);
                    atomicAdd(&out[OFF_COLOR + 3 * p + 2], w * c2v);
                    atomicAdd(&out[OFF_IMW + p], w * w0v);           // imw
                }
            }
        }
    }
}

// ---------------------------------------------------------------------------
extern "C" void kernel_launch(void* const* d_in, const int* in_sizes, int n_in,
                              void* d_out, int out_size, void* d_ws, size_t ws_size,
                              hipStream_t stream) {
    const float* pts  = (const float*)d_in[0];   // [B,N,3]
    const float* col  = (const float*)d_in[1];   // [B,N,3]
    const float* imw  = (const float*)d_in[2];   // [B,N,1]
    const int*   mask = (const int*)d_in[3];     // [B,N]
    const float* th   = (const float*)d_in[4];   // [1]
    float* out = (float*)d_out;

    int N = in_sizes[3];                         // B*N (B==1)

    float* dws  = (float*)d_ws;                  // depth      [HW]
    float* rmin = dws + HWSZ;                    // row min    [HW]
    float* pmin = rmin + HWSZ;                   // patch min  [HW]

    const int T = 256;                           // 8 wave32 waves / block
    k_init      <<<(5 * HWSZ + T - 1) / T, T, 0, stream>>>(dws, out);
    k_zmin      <<<(N + T - 1) / T,        T, 0, stream>>>(pts, mask,
                                                           (unsigned int*)dws, N);
    k_rowmin    <<<(HWSZ + T - 1) / T,     T, 0, stream>>>(dws, rmin, out);
    k_colmin_tdm<<<(WW / TS_X) * (HH / TS_Y), T, 0, stream>>>(rmin, pmin);
    k_splat_wmma<<<(N + T - 1) / T,        T, 0, stream>>>(pts, col, imw, mask,
                                                           th, pmin, out, N);
}